// NGCF_27711128994652
// MI455X (gfx1250) — compile-verified
//
#include <hip/hip_runtime.h>
#include <hip/hip_bf16.h>
#include <math.h>

#define N_USER 50000
#define N_ITEM 50000
#define N_NODES (N_USER + N_ITEM)      // 100000, divisible by 16
#define NNZ 3200000
#define EMB 64
#define BATCH 4096
#define NEG_SLOPE 0.2f

typedef __attribute__((ext_vector_type(2))) float v2f;
typedef __attribute__((ext_vector_type(8))) float v8f;

// ---------------------------------------------------------------------------
// ego = concat(user_emb, item_emb)   (N_NODES x 64), float4 copies
// ---------------------------------------------------------------------------
__global__ void ngcf_concat(const float* __restrict__ ue,
                            const float* __restrict__ ie,
                            float* __restrict__ ego) {
    int gid = blockIdx.x * blockDim.x + threadIdx.x;   // over N_NODES*16 float4
    if (gid >= N_NODES * 16) return;
    float4 v;
    if (gid < N_USER * 16) v = ((const float4*)ue)[gid];
    else                   v = ((const float4*)ie)[gid - N_USER * 16];
    ((float4*)ego)[gid] = v;
}

// ---------------------------------------------------------------------------
// SpMM scatter: side[row] += val * ego[col].  16 threads per edge, 4 floats
// each (float4 load + 4 global_atomic_add_f32, all L2-resident).
// ---------------------------------------------------------------------------
__global__ void ngcf_spmm(const int* __restrict__ arow,
                          const int* __restrict__ acol,
                          const float* __restrict__ aval,
                          const float* __restrict__ x,
                          float* __restrict__ y) {
    unsigned gid = blockIdx.x * blockDim.x + threadIdx.x;  // NNZ*16 threads
    unsigned e = gid >> 4;
    unsigned part = gid & 15u;
    if (e >= NNZ) return;
    int r = arow[e];
    int c = acol[e];
    float v = aval[e];
    float4 xv = *(const float4*)(x + (size_t)c * EMB + part * 4);
    float* yp = y + (size_t)r * EMB + part * 4;
    atomicAdd(yp + 0, v * xv.x);
    atomicAdd(yp + 1, v * xv.y);
    atomicAdd(yp + 2, v * xv.z);
    atomicAdd(yp + 3, v * xv.w);
}

// ---------------------------------------------------------------------------
// Layer kernel: ego_new = leaky_relu(side@Wgc + bgc + (ego*side)@Wbi + bbi)
// FP32 WMMA 16x16x4. Block = 128 threads (4 waves); block owns 16 rows,
// wave w owns N-columns [16w, 16w+16). Both GEMMs fused into one accumulator:
// 32 v_wmma_f32_16x16x4_f32 per wave. In-place ego update after barrier.
// ---------------------------------------------------------------------------
__global__ void ngcf_layer(const float* __restrict__ side,
                           float* __restrict__ ego,
                           const float* __restrict__ Wgc,
                           const float* __restrict__ bgc,
                           const float* __restrict__ Wbi,
                           const float* __restrict__ bbi) {
    const int lane  = threadIdx.x & 31;
    const int wv    = threadIdx.x >> 5;          // 0..3 -> N tile
    const int mbase = blockIdx.x * 16;
    const int m     = lane & 15;                 // A row within tile
    const int khalf = (lane >> 4) << 1;          // 0 (lanes 0-15) or 2 (16-31)
    const int n     = (lane & 15) + wv * 16;     // B/D column

    const float* __restrict__ Srow = side + (size_t)(mbase + m) * EMB;
    const float* __restrict__ Erow = ego  + (size_t)(mbase + m) * EMB;

    v8f acc = {};

    // side @ Wgc  (16 K-steps of 4)
#pragma unroll
    for (int k0 = 0; k0 < EMB; k0 += 4) {
        const int kk = k0 + khalf;
        v2f a, b;
        a.x = Srow[kk];
        a.y = Srow[kk + 1];
        b.x = Wgc[kk * EMB + n];
        b.y = Wgc[(kk + 1) * EMB + n];
        acc = __builtin_amdgcn_wmma_f32_16x16x4_f32(
            false, a, false, b, (short)0, acc, false, false);
    }
    // (ego * side) @ Wbi, same accumulator
#pragma unroll
    for (int k0 = 0; k0 < EMB; k0 += 4) {
        const int kk = k0 + khalf;
        v2f a, b;
        a.x = Srow[kk]     * Erow[kk];
        a.y = Srow[kk + 1] * Erow[kk + 1];
        b.x = Wbi[kk * EMB + n];
        b.y = Wbi[(kk + 1) * EMB + n];
        acc = __builtin_amdgcn_wmma_f32_16x16x4_f32(
            false, a, false, b, (short)0, acc, false, false);
    }

    const float bias = bgc[n] + bbi[n];
    float vals[8];
#pragma unroll
    for (int r = 0; r < 8; ++r) {
        float v = acc[r] + bias;
        vals[r] = (v > 0.0f) ? v : NEG_SLOPE * v;
    }

    __syncthreads();   // all reads of ego/side done before in-place writes

    // D layout: VGPR r -> row = r (lanes 0-15) / r+8 (lanes 16-31), col = lane&15
    const int rowoff = (lane >> 4) << 3;
#pragma unroll
    for (int r = 0; r < 8; ++r) {
        ego[(size_t)(mbase + r + rowoff) * EMB + n] = vals[r];
    }
}

// ---------------------------------------------------------------------------
// Gather one 64-col slice of the output for users/pos/neg.
// blockIdx: [0,3*4096), 64 threads = one row. Optional row L2 normalization
// (recomputed on the fly; next-layer ego stays un-normalized per reference).
// d_out layout: [3][4096][256] floats, slice*64 column offset.
// ---------------------------------------------------------------------------
__global__ void ngcf_gather(const int* __restrict__ users,
                            const int* __restrict__ pos,
                            const int* __restrict__ neg,
                            const float* __restrict__ ego,
                            float* __restrict__ out,
                            int slice, int do_norm) {
    __shared__ float sm[64];
    const int w = blockIdx.x >> 12;         // 0=users, 1=pos, 2=neg
    const int b = blockIdx.x & 4095;
    const int t = threadIdx.x;              // 0..63

    int idx;
    if (w == 0)      idx = users[b];
    else if (w == 1) idx = N_USER + pos[b];
    else             idx = N_USER + neg[b];

    float v = ego[(size_t)idx * EMB + t];
    float scale = 1.0f;
    if (do_norm) {
        sm[t] = v * v;
        __syncthreads();
        float s = 0.0f;
#pragma unroll
        for (int i = 0; i < 64; ++i) s += sm[i];
        float nrm = fmaxf(sqrtf(s), 1e-12f);
        scale = 1.0f / nrm;
    }
    out[((size_t)w * BATCH + b) * 256 + slice * 64 + t] = v * scale;
}

// ---------------------------------------------------------------------------
extern "C" void kernel_launch(void* const* d_in, const int* in_sizes, int n_in,
                              void* d_out, int out_size, void* d_ws, size_t ws_size,
                              hipStream_t stream) {
    const int*   users  = (const int*)d_in[0];
    const int*   pos    = (const int*)d_in[1];
    const int*   neg    = (const int*)d_in[2];
    const int*   arow   = (const int*)d_in[3];
    const int*   acol   = (const int*)d_in[4];
    const float* aval   = (const float*)d_in[5];
    const float* uemb   = (const float*)d_in[6];
    const float* iemb   = (const float*)d_in[7];
    const float* W_gc   = (const float*)d_in[8];
    const float* b_gc   = (const float*)d_in[9];
    const float* W_bi   = (const float*)d_in[10];
    const float* b_bi   = (const float*)d_in[11];
    float* out = (float*)d_out;

    // workspace: ego (N x 64 f32) | side (N x 64 f32)  => 51.2 MB
    float* ego  = (float*)d_ws;
    float* side = ego + (size_t)N_NODES * EMB;

    // ego0 = concat
    {
        int total = N_NODES * 16;
        ngcf_concat<<<(total + 255) / 256, 256, 0, stream>>>(uemb, iemb, ego);
    }
    // layer-0 slice of output (raw embeddings, no norm)
    ngcf_gather<<<3 * BATCH, 64, 0, stream>>>(users, pos, neg, ego, out, 0, 0);

    for (int k = 0; k < 3; ++k) {
        hipMemsetAsync(side, 0, (size_t)N_NODES * EMB * sizeof(float), stream);

        {
            unsigned total = (unsigned)NNZ * 16u;
            ngcf_spmm<<<(total + 255u) / 256u, 256, 0, stream>>>(
                arow, acol, aval, ego, side);
        }

        ngcf_layer<<<N_NODES / 16, 128, 0, stream>>>(
            side, ego,
            W_gc + (size_t)k * EMB * EMB, b_gc + (size_t)k * EMB,
            W_bi + (size_t)k * EMB * EMB, b_bi + (size_t)k * EMB);

        // slice k+1 of output: normalized rows, gathered straight into d_out
        ngcf_gather<<<3 * BATCH, 64, 0, stream>>>(
            users, pos, neg, ego, out, k + 1, 1);
    }
    (void)in_sizes; (void)n_in; (void)out_size; (void)ws_size;
}